// Generator_19078244729188
// MI455X (gfx1250) — compile-verified
//
#include <hip/hip_runtime.h>

typedef __attribute__((ext_vector_type(2))) float v2f;
typedef __attribute__((ext_vector_type(8))) float v8f;

#define H_DIM  256
#define S_SAMP 32
#define PXPX   4096
#define H0_STRIDE 260   // 256 + 4 pad -> conflict-free ds_load_b64 A-fragments

// ---------------------------------------------------------------------------
// Fill output with background = 1.0
// ---------------------------------------------------------------------------
__global__ void fill_kernel(float* __restrict__ out, int n) {
  int i = blockIdx.x * 256 + threadIdx.x;
  if (i < n) out[i] = 1.0f;
}

// ---------------------------------------------------------------------------
// Per-batch setup: latent bias  latb[b][h] = b0[h] + sum_l lat[b][l]*W0[3+l][h]
// plus rotation matrix R[b] (row-major 3x3) and camera location cam[b].
// ---------------------------------------------------------------------------
__global__ __launch_bounds__(256) void setup_kernel(
    const float* __restrict__ latents, const float* __restrict__ phis,
    const float* __restrict__ thetas,  const float* __restrict__ W0,
    const float* __restrict__ b0,
    float* __restrict__ latb_ws, float* __restrict__ R_ws, float* __restrict__ cam_ws,
    int L)
{
  const int b = blockIdx.x;
  const int h = threadIdx.x;
  float acc = b0[h];
  for (int l = 0; l < L; ++l)
    acc = fmaf(latents[b * L + l], W0[(3 + l) * H_DIM + h], acc);
  latb_ws[b * H_DIM + h] = acc;

  if (h == 0) {
    const float cp = cosf(phis[b]),  sp = sinf(phis[b]);
    const float ct = cosf(thetas[b]), st = sinf(thetas[b]);
    // columns: right=[ct,0,-st], up=[-sp*st,cp,-sp*ct], back=[cp*st,sp,cp*ct]
    float R[9];
    R[0] = ct;   R[1] = -sp * st; R[2] = cp * st;
    R[3] = 0.f;  R[4] = cp;       R[5] = sp;
    R[6] = -st;  R[7] = -sp * ct; R[8] = cp * ct;
    for (int i = 0; i < 9; ++i) R_ws[b * 9 + i] = R[i];
    cam_ws[b * 3 + 0] = 2.0f * R[2];
    cam_ws[b * 3 + 1] = 2.0f * R[5];
    cam_ws[b * 3 + 2] = 2.0f * R[8];
  }
}

// ---------------------------------------------------------------------------
// Repack W1 [256x256 row-major] into per-WMMA-step B-fragment layout:
// W1p[((kk*16 + nt)*32 + lane)*2 + j] = W1[(4*kk + 2*(lane>>4) + j)*256 + nt*16 + (lane&15)]
// ---------------------------------------------------------------------------
__global__ void repack_kernel(const float* __restrict__ W1, float* __restrict__ W1p) {
  const int tid  = blockIdx.x * 256 + threadIdx.x;   // 0..32767
  const int lane = tid & 31;
  const int nt   = (tid >> 5) & 15;
  const int kk   = tid >> 9;
  const int n    = nt * 16 + (lane & 15);
  const int k    = 4 * kk + 2 * (lane >> 4);
  W1p[tid * 2 + 0] = W1[(k + 0) * H_DIM + n];
  W1p[tid * 2 + 1] = W1[(k + 1) * H_DIM + n];
}

// ---------------------------------------------------------------------------
// Main kernel: one 32-thread workgroup (one wave) per ray. The wave's 32
// samples form two WMMA M-tiles that SHARE every B-fragment load (2x W1
// reuse, 2 independent v_wmma accumulation chains). Layer0 (3x256) on VALU
// into LDS, layer1 (256x256) via v_wmma_f32_16x16x4_f32, layer2 + min fused.
// ---------------------------------------------------------------------------
__global__ __launch_bounds__(32) void render_kernel(
    const float* __restrict__ viewField,
    const int*   __restrict__ mask_idx,
    const float* __restrict__ nearv,
    const float* __restrict__ farv,
    const float* __restrict__ W0,
    const float* __restrict__ W1,
    const float* __restrict__ b1,
    const float* __restrict__ W2,
    const float* __restrict__ b2,
    const float* __restrict__ latb_ws,
    const float* __restrict__ R_ws,
    const float* __restrict__ cam_ws,
    const float* __restrict__ W1p,
    int use_packed,
    float* __restrict__ out,
    int M)
{
  __shared__ float h0s[S_SAMP * H0_STRIDE + 4];

  const int lane = threadIdx.x;    // 0..31, one wave
  const int l16  = lane & 15;
  const int hl   = lane >> 4;      // lane half (K offset inside fragments)

  const int rid  = blockIdx.x;
  const int b    = rid / M;
  const int mray = rid - b * M;

  // --- wave-uniform ray setup -------------------------------------------
  const int pix = mask_idx[mray];
  const float dx = viewField[pix * 3 + 0];
  const float dy = viewField[pix * 3 + 1];
  const float dz = viewField[pix * 3 + 2];

  const float* Rb = R_ws + b * 9;
  const float e0 = Rb[0] * dx + Rb[1] * dy + Rb[2] * dz;   // world ray dir
  const float e1 = Rb[3] * dx + Rb[4] * dy + Rb[5] * dz;
  const float e2 = Rb[6] * dx + Rb[7] * dy + Rb[8] * dz;
  const float cx = cam_ws[b * 3 + 0];
  const float cy = cam_ws[b * 3 + 1];
  const float cz = cam_ws[b * 3 + 2];

  const float nr = nearv[mray];
  const float dt = (farv[mray] - nr) * (1.0f / (float)(S_SAMP - 1));

  const float* latb = latb_ws + b * H_DIM;

  // --- layer 0: h0[s][k] = relu(p(s) . W0[0:3,k] + latb[k]), s = 0..31 ---
  for (int kb = 0; kb < 8; ++kb) {
    const int k = kb * 32 + lane;
    const float w0x = W0[0 * H_DIM + k];
    const float w0y = W0[1 * H_DIM + k];
    const float w0z = W0[2 * H_DIM + k];
    const float lb  = latb[k];
    #pragma unroll
    for (int m = 0; m < S_SAMP; ++m) {
      const float t  = fmaf(dt, (float)m, nr);
      const float px = fmaf(e0, t, cx);
      const float py = fmaf(e1, t, cy);
      const float pz = fmaf(e2, t, cz);
      const float v  = fmaf(px, w0x, fmaf(py, w0y, fmaf(pz, w0z, lb)));
      h0s[m * H0_STRIDE + k] = fmaxf(v, 0.0f);
    }
  }
  __syncthreads();   // single-wave WG: hardware NOP, acts as LDS fence

  // --- layer 1 (WMMA f32 16x16x4, two M-tiles per B fragment) -----------
  float sacc0[8] = {0, 0, 0, 0, 0, 0, 0, 0};
  float sacc1[8] = {0, 0, 0, 0, 0, 0, 0, 0};
  const float* a0base = &h0s[l16 * H0_STRIDE + 2 * hl];      // samples 0..15
  const float* a1base = a0base + 16 * H0_STRIDE;             // samples 16..31

  if (use_packed) {
    const v2f* W1v = (const v2f*)W1p;
    for (int nt = 0; nt < 16; ++nt) {
      v8f acc0 = {0, 0, 0, 0, 0, 0, 0, 0};
      v8f acc1 = {0, 0, 0, 0, 0, 0, 0, 0};
      #pragma unroll 4
      for (int kk = 0; kk < 64; ++kk) {
        v2f bv = W1v[(kk * 16 + nt) * 32 + lane];
        v2f a0 = *(const v2f*)(a0base + 4 * kk);
        v2f a1 = *(const v2f*)(a1base + 4 * kk);
        acc0 = __builtin_amdgcn_wmma_f32_16x16x4_f32(false, a0, false, bv,
                                                     (short)0, acc0, false, false);
        acc1 = __builtin_amdgcn_wmma_f32_16x16x4_f32(false, a1, false, bv,
                                                     (short)0, acc1, false, false);
      }
      const int n = nt * 16 + l16;
      const float bl  = b1[n];
      const float w2v = W2[n];
      #pragma unroll
      for (int r = 0; r < 8; ++r) {
        float v0 = fmaxf(acc0[r] + bl, 0.0f);
        float v1 = fmaxf(acc1[r] + bl, 0.0f);
        sacc0[r] = fmaf(v0, w2v, sacc0[r]);
        sacc1[r] = fmaf(v1, w2v, sacc1[r]);
      }
    }
  } else {
    for (int nt = 0; nt < 16; ++nt) {
      const int n = nt * 16 + l16;
      const float* w1c = W1 + n;
      v8f acc0 = {0, 0, 0, 0, 0, 0, 0, 0};
      v8f acc1 = {0, 0, 0, 0, 0, 0, 0, 0};
      #pragma unroll 4
      for (int kk = 0; kk < 64; ++kk) {
        const int k = 4 * kk + 2 * hl;
        v2f bv;
        bv.x = w1c[(k + 0) * H_DIM];
        bv.y = w1c[(k + 1) * H_DIM];
        v2f a0 = *(const v2f*)(a0base + 4 * kk);
        v2f a1 = *(const v2f*)(a1base + 4 * kk);
        acc0 = __builtin_amdgcn_wmma_f32_16x16x4_f32(false, a0, false, bv,
                                                     (short)0, acc0, false, false);
        acc1 = __builtin_amdgcn_wmma_f32_16x16x4_f32(false, a1, false, bv,
                                                     (short)0, acc1, false, false);
      }
      const float bl  = b1[n];
      const float w2v = W2[n];
      #pragma unroll
      for (int r = 0; r < 8; ++r) {
        float v0 = fmaxf(acc0[r] + bl, 0.0f);
        float v1 = fmaxf(acc1[r] + bl, 0.0f);
        sacc0[r] = fmaf(v0, w2v, sacc0[r]);
        sacc1[r] = fmaf(v1, w2v, sacc1[r]);
      }
    }
  }

  // --- reduce: sum each row over the 16 lanes of its half (covers all
  //     256 n), add b2, min over the 16 rows, then across halves. --------
  const float b2v = b2[0];
  float m8 = 3.402823466e+38f;
  #pragma unroll
  for (int r = 0; r < 8; ++r) {
    float s0 = sacc0[r];
    s0 += __shfl_xor(s0, 1);
    s0 += __shfl_xor(s0, 2);
    s0 += __shfl_xor(s0, 4);
    s0 += __shfl_xor(s0, 8);
    m8 = fminf(m8, s0 + b2v);
    float s1 = sacc1[r];
    s1 += __shfl_xor(s1, 1);
    s1 += __shfl_xor(s1, 2);
    s1 += __shfl_xor(s1, 4);
    s1 += __shfl_xor(s1, 8);
    m8 = fminf(m8, s1 + b2v);
  }
  m8 = fminf(m8, __shfl_xor(m8, 16));   // min over all 32 samples of the ray

  if (lane == 0) {
    float shade = m8 * 20.0f;                   // SHADE_SCALE
    shade = fminf(fmaxf(shade, 0.0f), 1.0f);
    out[b * PXPX + pix] = shade;
  }
}

// ---------------------------------------------------------------------------
extern "C" void kernel_launch(void* const* d_in, const int* in_sizes, int n_in,
                              void* d_out, int out_size, void* d_ws, size_t ws_size,
                              hipStream_t stream) {
  const float* latents   = (const float*)d_in[0];
  const float* phis      = (const float*)d_in[1];
  const float* thetas    = (const float*)d_in[2];
  const float* viewField = (const float*)d_in[3];
  const float* nearv     = (const float*)d_in[4];
  const float* farv      = (const float*)d_in[5];
  const float* W0        = (const float*)d_in[6];
  const float* b0        = (const float*)d_in[7];
  const float* W1        = (const float*)d_in[8];
  const float* b1        = (const float*)d_in[9];
  const float* W2        = (const float*)d_in[10];
  const float* b2        = (const float*)d_in[11];
  const int*   mask_idx  = (const int*)d_in[12];

  const int Bn = in_sizes[1];            // batch
  const int L  = in_sizes[0] / Bn;       // latent dim
  const int M  = in_sizes[12];           // masked ray count

  float* ws      = (float*)d_ws;
  float* latb_ws = ws;                   // Bn*256
  float* R_ws    = latb_ws + Bn * H_DIM; // Bn*9
  float* cam_ws  = R_ws + Bn * 9;        // Bn*3
  int base = Bn * H_DIM + Bn * 9 + Bn * 3;
  base = (base + 15) & ~15;
  float* W1p = ws + base;                // 256*256 repacked fragments
  const size_t needed = (size_t)(base + H_DIM * H_DIM) * sizeof(float);
  const int use_packed = (ws_size >= needed) ? 1 : 0;

  float* out = (float*)d_out;

  fill_kernel<<<(out_size + 255) / 256, 256, 0, stream>>>(out, out_size);
  setup_kernel<<<Bn, 256, 0, stream>>>(latents, phis, thetas, W0, b0,
                                       latb_ws, R_ws, cam_ws, L);
  if (use_packed)
    repack_kernel<<<(H_DIM * H_DIM / 2) / 256, 256, 0, stream>>>(W1, W1p);

  render_kernel<<<Bn * M, 32, 0, stream>>>(
      viewField, mask_idx, nearv, farv, W0, W1, b1, W2, b2,
      latb_ws, R_ws, cam_ws, W1p, use_packed, out, M);
}